// CircleLossV2_23553600651800
// MI455X (gfx1250) — compile-verified
//
#include <hip/hip_runtime.h>
#include <hip/hip_bf16.h>

typedef __attribute__((ext_vector_type(2))) float v2f;
typedef __attribute__((ext_vector_type(8))) float v8f;

#define B_SZ 8192
#define D_SZ 128
#define PANEL 64              // columns staged in LDS per iteration
#define CSTRIDE 132           // padded floats per column (132%64=4 -> conflict-free)
#define ROWS_PER_WAVE 16
#define WAVES_PER_BLOCK 4
#define ROWS_PER_BLOCK (ROWS_PER_WAVE * WAVES_PER_BLOCK)
#define NSPLIT 8              // column chunks (parallelism for ~1000 SIMDs)
#define COLS_PER_CHUNK (B_SZ / NSPLIT)

#define GAMMA_C 64.0f
#define O_P 0.75f             // 1 - margin
#define O_N (-0.75f)          // -1 + margin
#define NEG_INF_C (-1.0e30f)

// ---------- kernel 0: zero the two global accumulators ----------
__global__ void circle_init(float* acc) {
  if (threadIdx.x < 2) acc[threadIdx.x] = 0.0f;
}

// ---------- kernel 1: row L2-normalize (one wave32 per row) ----------
__global__ __launch_bounds__(256) void circle_normalize(const float* __restrict__ x,
                                                        float* __restrict__ en) {
  const int lane = threadIdx.x & 31;
  const int wave = threadIdx.x >> 5;
  const int row  = blockIdx.x * 8 + wave;
  if (row >= B_SZ) return;
  const float4 v = *(const float4*)(x + (size_t)row * D_SZ + lane * 4);
  float ss = v.x * v.x + v.y * v.y + v.z * v.z + v.w * v.w;
  ss += __shfl_xor(ss, 1, 32);
  ss += __shfl_xor(ss, 2, 32);
  ss += __shfl_xor(ss, 4, 32);
  ss += __shfl_xor(ss, 8, 32);
  ss += __shfl_xor(ss, 16, 32);
  const float r = rsqrtf(ss);
  float4 o;
  o.x = v.x * r; o.y = v.y * r; o.z = v.z * r; o.w = v.w * r;
  *(float4*)(en + (size_t)row * D_SZ + lane * 4) = o;
}

// ---------- kernel 2: fused normalized-GEMM + masked-LSE partials ----------
// blockIdx.x: row block (64 rows), blockIdx.y: column chunk (1024 cols).
// Each lane keeps a private online-LSE state per row slot; single exp per
// update. Per-row partial (m_p, s_p, m_n, s_n) is written to workspace.
__global__ __launch_bounds__(WAVES_PER_BLOCK * 32) void circle_main(
    const float* __restrict__ en, const int* __restrict__ labels,
    float* __restrict__ partial) {
  __shared__ float sB[PANEL * CSTRIDE];
  __shared__ int   sLab[PANEL];

  const int tid  = threadIdx.x;
  const int lane = tid & 31;
  const int wave = tid >> 5;
  const int hi   = lane >> 4;      // 0: lanes 0-15, 1: lanes 16-31
  const int l16  = lane & 15;

  const int i0w    = blockIdx.x * ROWS_PER_BLOCK + wave * ROWS_PER_WAVE;
  const int chunk  = blockIdx.y;
  const int colLo  = chunk * COLS_PER_CHUNK;

  // ---- A fragments for this wave's 16 rows, full K=128 (kept in VGPRs) ----
  // fp32 16x4 A layout: lanes 0-15 hold K = 4k..4k+1, lanes 16-31 K = 4k+2..4k+3
  v2f a[32];
  {
    const float* arow = en + (size_t)(i0w + l16) * D_SZ + hi * 2;
#pragma unroll
    for (int ks = 0; ks < 32; ++ks) {
      a[ks].x = arow[ks * 4 + 0];
      a[ks].y = arow[ks * 4 + 1];
    }
  }

  // row labels: slot r (r=0..7) -> row i0w + hi*8 + r (replicated over 16 lanes)
  int rl[8];
#pragma unroll
  for (int r = 0; r < 8; ++r) rl[r] = labels[i0w + hi * 8 + r];

  // per-lane online LSE state per row slot
  float mp[8], sp[8], mn[8], sn[8];
#pragma unroll
  for (int r = 0; r < 8; ++r) { mp[r] = NEG_INF_C; mn[r] = NEG_INF_C; sp[r] = 0.f; sn[r] = 0.f; }

  for (int panel = 0; panel < COLS_PER_CHUNK / PANEL; ++panel) {
    const int j0 = colLo + panel * PANEL;
    __syncthreads();
    // cooperative stage: 64 cols x 128 floats, float4 chunks
    for (int idx = tid; idx < PANEL * (D_SZ / 4); idx += WAVES_PER_BLOCK * 32) {
      const int c  = idx >> 5;          // / (D/4)
      const int kk = (idx & 31) << 2;
      const float4 v = *(const float4*)(en + (size_t)(j0 + c) * D_SZ + kk);
      *(float4*)(&sB[c * CSTRIDE + kk]) = v;
    }
    if (tid < PANEL) sLab[tid] = labels[j0 + tid];
    __syncthreads();

#pragma unroll 1
    for (int ct = 0; ct < PANEL / 16; ++ct) {
      const int cbase = ct * 16;
      v8f c = {0.f, 0.f, 0.f, 0.f, 0.f, 0.f, 0.f, 0.f};
      const float* bcol = &sB[(cbase + l16) * CSTRIDE + hi * 2];
#pragma unroll
      for (int ks = 0; ks < 32; ++ks) {
        v2f b;
        b.x = bcol[ks * 4 + 0];
        b.y = bcol[ks * 4 + 1];
        c = __builtin_amdgcn_wmma_f32_16x16x4_f32(
            false, a[ks], false, b, (short)0, c, false, false);
      }

      const int colLab = sLab[cbase + l16];
      const int col    = j0 + cbase + l16;
#pragma unroll
      for (int r = 0; r < 8; ++r) {
        const float s   = c[r];
        const int   row = i0w + hi * 8 + r;
        const bool same = (rl[r] == colLab);
        const bool offd = (row != col);
        const float ap = fmaxf(O_P - s, 0.f);
        const float an = fmaxf(s - O_N, 0.f);
        const float lp = (same && offd)  ? (-GAMMA_C * ap * (s - O_P)) : NEG_INF_C;
        const float ln = (!same && offd) ? ( GAMMA_C * an * (s - O_N)) : NEG_INF_C;

        // single-exp branchless online LSE update
        {
          const float d  = lp - mp[r];
          const float e  = __expf(-fabsf(d));
          const bool  up = d > 0.f;
          sp[r] = up ? (sp[r] * e + 1.f) : (sp[r] + e);
          mp[r] = up ? lp : mp[r];
        }
        {
          const float d  = ln - mn[r];
          const float e  = __expf(-fabsf(d));
          const bool  up = d > 0.f;
          sn[r] = up ? (sn[r] * e + 1.f) : (sn[r] + e);
          mn[r] = up ? ln : mn[r];
        }
      }
    }
  }

  // ---- one-time cross-lane merge of (m, s) over the 16 lanes of each half ----
#pragma unroll
  for (int r = 0; r < 8; ++r) {
#pragma unroll
    for (int xm = 1; xm <= 8; xm <<= 1) {
      float om = __shfl_xor(mp[r], xm, 32);
      float os = __shfl_xor(sp[r], xm, 32);
      float M  = fmaxf(mp[r], om);
      sp[r] = sp[r] * __expf(mp[r] - M) + os * __expf(om - M);
      mp[r] = M;
      om = __shfl_xor(mn[r], xm, 32);
      os = __shfl_xor(sn[r], xm, 32);
      M  = fmaxf(mn[r], om);
      sn[r] = sn[r] * __expf(mn[r] - M) + os * __expf(om - M);
      mn[r] = M;
    }
  }

  // lane 0 / lane 16 store partial states for their half's 8 rows
  if (l16 == 0) {
#pragma unroll
    for (int r = 0; r < 8; ++r) {
      const int row = i0w + hi * 8 + r;
      float4 st;
      st.x = mp[r]; st.y = sp[r]; st.z = mn[r]; st.w = sn[r];
      *(float4*)(partial + ((size_t)chunk * B_SZ + row) * 4) = st;
    }
  }
}

// ---------- kernel 3: merge chunk partials per row, accumulate loss ----------
__global__ __launch_bounds__(256) void circle_merge(const float* __restrict__ partial,
                                                    float* __restrict__ acc) {
  const int row = blockIdx.x * 256 + threadIdx.x;
  float mp = NEG_INF_C, sp = 0.f, mn = NEG_INF_C, sn = 0.f;
#pragma unroll
  for (int c = 0; c < NSPLIT; ++c) {
    const float4 st = *(const float4*)(partial + ((size_t)c * B_SZ + row) * 4);
    float M = fmaxf(mp, st.x);
    sp = sp * __expf(mp - M) + st.y * __expf(st.x - M);
    mp = M;
    M = fmaxf(mn, st.z);
    sn = sn * __expf(mn - M) + st.w * __expf(st.z - M);
    mn = M;
  }
  // real logits are >= 0; masked sentinel -1e30 => validity via running max
  const bool valid = (mp > -1.0e29f) && (mn > -1.0e29f);
  float local = 0.f, cnt = 0.f;
  if (valid) {
    const float x = (mp + logf(sp)) + (mn + logf(sn));
    local = fmaxf(x, 0.f) + log1pf(expf(-fabsf(x)));  // softplus
    cnt = 1.f;
  }
  // wave-level reduction, then one atomic per wave
  for (int xm = 1; xm <= 16; xm <<= 1) {
    local += __shfl_xor(local, xm, 32);
    cnt   += __shfl_xor(cnt, xm, 32);
  }
  if ((threadIdx.x & 31) == 0) {
    atomicAdd(&acc[0], local);
    atomicAdd(&acc[1], cnt);
  }
}

// ---------- kernel 4: finalize scalar ----------
__global__ void circle_finalize(const float* acc, float* out) {
  if (threadIdx.x == 0 && blockIdx.x == 0) {
    out[0] = acc[0] / fmaxf(acc[1], 1.0f);
  }
}

extern "C" void kernel_launch(void* const* d_in, const int* in_sizes, int n_in,
                              void* d_out, int out_size, void* d_ws, size_t ws_size,
                              hipStream_t stream) {
  const float* emb    = (const float*)d_in[0];
  const int*   labels = (const int*)d_in[1];
  float* out = (float*)d_out;

  float* ws      = (float*)d_ws;
  float* acc     = ws;                          // 2 floats
  float* en      = ws + 16;                     // normalized embeddings (4MB)
  float* partial = en + (size_t)B_SZ * D_SZ;    // NSPLIT*B_SZ*4 floats (1MB)

  circle_init<<<1, 64, 0, stream>>>(acc);
  circle_normalize<<<B_SZ / 8, 256, 0, stream>>>(emb, en);
  dim3 grid(B_SZ / ROWS_PER_BLOCK, NSPLIT);
  circle_main<<<grid, WAVES_PER_BLOCK * 32, 0, stream>>>(en, labels, partial);
  circle_merge<<<B_SZ / 256, 256, 0, stream>>>(partial, acc);
  circle_finalize<<<1, 1, 0, stream>>>(acc, out);
}